// embedding_2731599200670
// MI455X (gfx1250) — compile-verified
//
#include <hip/hip_runtime.h>
#include <stdint.h>

// Embedding lookup: out[t, :] = weight[:, x[t]] + bias
//   x: [32768] int32, weight: [512, 50257] fp32, bias: [512] fp32
// Pass 1: Wt[v, e] = W[e, v] + bias[e]   (LDS-tiled transpose, bias fused)
// Pass 2: out[16k..16k+15, :] = Wt[x[16k..16k+15], :]  via TDM gather-mode DMA

#define VOCAB 50257
#define EMBED 512

typedef float float4v __attribute__((ext_vector_type(4)));
typedef unsigned int v4u __attribute__((ext_vector_type(4)));
typedef int v4i __attribute__((ext_vector_type(4)));
typedef int v8i __attribute__((ext_vector_type(8)));

// ---------------------------------------------------------------------------
// Pass 1: transpose + bias fuse.
// Load side: W rows have odd stride (50257) -> scalar b32 NT loads (aligned 4B).
// Store side: Wt rows are 2KB aligned -> b128 stores.
// LDS tile [32][33]: load-phase writes hit consecutive banks; store-phase
// reads tile[eq*4+k][v_l] -> bank (4*eq + v_l + 33k) mod 64, distinct per wave.
// ---------------------------------------------------------------------------
__global__ __launch_bounds__(256) void embed_transpose_bias_kernel(
    const float* __restrict__ W, const float* __restrict__ bias,
    float* __restrict__ Wt) {
  __shared__ float tile[32][33];

  const int tid = threadIdx.x;
  const int v0 = blockIdx.x * 32;
  const int e0 = blockIdx.y * 32;

  // Load phase: lane layout (tx = v within tile, ty = e within tile)
  {
    const int tx = tid & 31;
    const int ty = tid >> 5;  // 0..7
#pragma unroll
    for (int j = 0; j < 32; j += 8) {
      const int v = v0 + tx;
      const int e = e0 + ty + j;
      if (v < VOCAB) {
        tile[ty + j][tx] =
            __builtin_nontemporal_load(&W[(size_t)e * VOCAB + v]);
      }
    }
  }
  __syncthreads();

  // Store phase: lane owns (row v_l, e-quad eq); writes one float4 of Wt.
  {
    const int v_l = tid >> 3;  // 0..31
    const int eq = tid & 7;    // 0..7
    const int v = v0 + v_l;
    if (v < VOCAB) {
      const int e = e0 + eq * 4;
      float4v val;
#pragma unroll
      for (int k = 0; k < 4; ++k) val[k] = tile[eq * 4 + k][v_l];
      const float4v b4 = *(const float4v*)&bias[e];
      *(float4v*)&Wt[(size_t)v * EMBED + e] = val + b4;
    }
  }
}

// ---------------------------------------------------------------------------
// Pass 2: TDM gather. One wave per 16 consecutive tokens.
//   tensor_load_to_lds (gather_mode, 16-bit indices): 16 rows x 2KB -> LDS
//   tensor_store_from_lds: contiguous 32KB LDS -> out (tokens are consecutive)
// Toolchain note: clang-23 lane exposes the 6-arg builtin; args 1..4 are the
// ISA D# groups 0..3, arg 5 (int32x8) has no ISA counterpart (VADDR4 slot is
// "unused/NULL") -> pass zeros, cpol last.
// ---------------------------------------------------------------------------
__global__ __launch_bounds__(32) void embed_gather_tdm_kernel(
    const int* __restrict__ x, const float* __restrict__ Wt,
    float* __restrict__ out) {
  __shared__ float stage[16 * EMBED];  // 32 KB staging tile
  const int t0 = blockIdx.x * 16;

  // Pack 16 row indices into 8 dwords of u16 pairs (VOCAB < 65536).
  unsigned idxw[8];
#pragma unroll
  for (int i = 0; i < 8; ++i) {
    const unsigned lo = (unsigned)x[t0 + 2 * i];
    const unsigned hi = (unsigned)x[t0 + 2 * i + 1];
    idxw[i] = (lo & 0xFFFFu) | (hi << 16);
  }

  const unsigned lds_base = (unsigned)(uintptr_t)&stage[0];
  const uint64_t src = (uint64_t)(uintptr_t)Wt;
  const uint64_t dst = (uint64_t)(uintptr_t)(out + (size_t)t0 * EMBED);

  // ---- D# group0/group1 for the gather load (see CDNA5 ISA) ----
  // g0: w0 = count=1 | gather_index_size=0(16b)<<30 | gather_mode=1<<31
  //     w1 = lds_addr; w2 = global_addr[31:0]; w3 = addr[56:32] | type=2<<30
  v4u g0l = {1u | (1u << 31), lds_base, (unsigned)(src & 0xFFFFFFFFu),
             (unsigned)((src >> 32) & 0x1FFFFFFu) | (2u << 30)};
  // g1: w0 = data_size=2(4B)<<16
  //     tensor_dim0=512 @bits[79:48]  -> w1[31:16]
  //     tensor_dim1=50257 @bits[111:80]-> w2[31:16] (low16, high16=0)
  //     tile_dim0=512 @bits[127:112]  -> w3[31:16]
  //     tile_dim1=#indices=16 @bits[143:128] -> w4[15:0]
  //     tensor_dim0_stride=512 @bits[207:160] -> w5
  v8i g1l = {0x20000,
             (int)((unsigned)EMBED << 16),
             (int)((unsigned)(VOCAB & 0xFFFF) << 16),
             (int)((unsigned)EMBED << 16),
             16,
             EMBED,
             0, 0};
  v4i g2l = {(int)idxw[0], (int)idxw[1], (int)idxw[2], (int)idxw[3]};
  v4i g3l = {(int)idxw[4], (int)idxw[5], (int)idxw[6], (int)idxw[7]};
  v8i gx = {0, 0, 0, 0, 0, 0, 0, 0};  // VADDR4 / reserved slot

  __builtin_amdgcn_tensor_load_to_lds(g0l, g1l, g2l, g3l, gx, 0);
  __builtin_amdgcn_s_wait_tensorcnt(0);

  // ---- D# for the store: 1-D contiguous tile of 8192 fp32, LDS -> out ----
  v4u g0s = {1u, lds_base, (unsigned)(dst & 0xFFFFFFFFu),
             (unsigned)((dst >> 32) & 0x1FFFFFFu) | (2u << 30)};
  v8i g1s = {0x20000,
             (int)((8192u & 0xFFFFu) << 16),  // tensor_dim0 = 8192
             (int)(1u << 16),                 // tensor_dim1 = 1
             (int)(0x2000u << 16),            // tile_dim0 = 8192
             0,                               // tile_dim1 = 0 (1-D tile)
             8192,                            // tensor_dim0_stride
             0, 0};
  v4i gz = {0, 0, 0, 0};

  __builtin_amdgcn_tensor_store_from_lds(g0s, g1s, gz, gz, gx, 0);
  __builtin_amdgcn_s_wait_tensorcnt(0);
}

// ---------------------------------------------------------------------------
// VALU tail (ntok % 16 tokens): pure row copy, bias already fused into Wt.
// ---------------------------------------------------------------------------
__global__ __launch_bounds__(128) void embed_gather_tail_kernel(
    const int* __restrict__ x, const float4v* __restrict__ Wt4,
    float4v* __restrict__ out4, int start, int ntok) {
  const int t = start + blockIdx.x;
  if (t >= ntok) return;
  const int v = x[t];
  const int tid = threadIdx.x;
  __builtin_nontemporal_store(Wt4[(size_t)v * (EMBED / 4) + tid],
                              &out4[(size_t)t * (EMBED / 4) + tid]);
}

// ---------------------------------------------------------------------------
// Fallback if workspace < 103 MB: direct strided gather (transaction-bound).
// ---------------------------------------------------------------------------
__global__ __launch_bounds__(256) void embed_direct_kernel(
    const int* __restrict__ x, const float* __restrict__ W,
    const float* __restrict__ bias, float* __restrict__ out, int ntok) {
  const int t = blockIdx.x;
  if (t >= ntok) return;
  const int e = blockIdx.y * 256 + threadIdx.x;
  const int v = x[t];
  out[(size_t)t * EMBED + e] = W[(size_t)e * VOCAB + v] + bias[e];
}

extern "C" void kernel_launch(void* const* d_in, const int* in_sizes, int n_in,
                              void* d_out, int out_size, void* d_ws,
                              size_t ws_size, hipStream_t stream) {
  const int* x = (const int*)d_in[0];
  const float* W = (const float*)d_in[1];
  const float* bias = (const float*)d_in[2];
  float* out = (float*)d_out;
  const int ntok = in_sizes[0];  // 32768

  const size_t wt_bytes = (size_t)VOCAB * EMBED * sizeof(float);  // ~103 MB

  if (ws_size >= wt_bytes) {
    float* Wt = (float*)d_ws;
    dim3 tb(256);
    dim3 tg((VOCAB + 31) / 32, EMBED / 32);
    embed_transpose_bias_kernel<<<tg, tb, 0, stream>>>(W, bias, Wt);

    const int ngrp = ntok / 16;
    if (ngrp > 0) {
      embed_gather_tdm_kernel<<<ngrp, 32, 0, stream>>>(x, Wt, out);
    }
    const int tail = ntok - ngrp * 16;
    if (tail > 0) {
      embed_gather_tail_kernel<<<tail, 128, 0, stream>>>(
          x, (const float4v*)Wt, (float4v*)out, ngrp * 16, ntok);
    }
  } else {
    dim3 g(ntok, EMBED / 256);
    embed_direct_kernel<<<g, 256, 0, stream>>>(x, W, bias, out, ntok);
  }
}